// MoELayer_74620761800731
// MI455X (gfx1250) — compile-verified
//
#include <hip/hip_runtime.h>
#include <math.h>

#define BB 8192
#define DD 2048
#define HH 4096
#define CC 1000
#define EE 8
#define KKE 2

typedef float v2f __attribute__((ext_vector_type(2)));
typedef float v8f __attribute__((ext_vector_type(8)));

__device__ __forceinline__ v8f wmma_f32(v2f a, v2f b, v8f c) {
  // V_WMMA_F32_16X16X4_F32 : D = A(16x4 f32) * B(4x16 f32) + C(16x16 f32)
  return __builtin_amdgcn_wmma_f32_16x16x4_f32(
      /*neg_a=*/false, a, /*neg_b=*/false, b,
      /*c_mod=*/(short)0, c, /*reuse_a=*/false, /*reuse_b=*/false);
}

// ---------------------------------------------------------------------------
// Kernel 1: gating. One wave (32 lanes) per token.
// ---------------------------------------------------------------------------
__global__ __launch_bounds__(256) void gate_kernel(
    const float* __restrict__ x, const float* __restrict__ Wg,
    const float* __restrict__ bg, float* __restrict__ gs_out,
    int* __restrict__ top_idx, float* __restrict__ top_w,
    int* __restrict__ counts) {
  const int wv = threadIdx.x >> 5;
  const int lane = threadIdx.x & 31;
  const int b = blockIdx.x * 8 + wv;
  if (b >= BB) return;

  float acc[EE];
#pragma unroll
  for (int e = 0; e < EE; ++e) acc[e] = 0.0f;

  const float* xr = x + (size_t)b * DD;
  for (int d = lane; d < DD; d += 32) {
    float xv = xr[d];
    const float4* g = (const float4*)(Wg + (size_t)d * EE);
    float4 g0 = g[0];
    float4 g1 = g[1];
    acc[0] += xv * g0.x; acc[1] += xv * g0.y;
    acc[2] += xv * g0.z; acc[3] += xv * g0.w;
    acc[4] += xv * g1.x; acc[5] += xv * g1.y;
    acc[6] += xv * g1.z; acc[7] += xv * g1.w;
  }
#pragma unroll
  for (int e = 0; e < EE; ++e) {
#pragma unroll
    for (int off = 16; off > 0; off >>= 1)
      acc[e] += __shfl_xor(acc[e], off, 32);
  }

  if (lane == 0) {
    float lg[EE];
#pragma unroll
    for (int e = 0; e < EE; ++e) lg[e] = acc[e] + bg[e];
    float mx = lg[0];
#pragma unroll
    for (int e = 1; e < EE; ++e) mx = fmaxf(mx, lg[e]);
    float ex[EE], se = 0.0f;
#pragma unroll
    for (int e = 0; e < EE; ++e) { ex[e] = expf(lg[e] - mx); se += ex[e]; }
    float inv = 1.0f / se;
    float gs[EE];
#pragma unroll
    for (int e = 0; e < EE; ++e) {
      gs[e] = ex[e] * inv;
      gs_out[(size_t)b * EE + e] = gs[e];
    }
    // top-2 (first occurrence wins on ties, like jax top_k)
    int i0 = 0;
#pragma unroll
    for (int e = 1; e < EE; ++e) if (gs[e] > gs[i0]) i0 = e;
    int i1 = (i0 == 0) ? 1 : 0;
#pragma unroll
    for (int e = 0; e < EE; ++e)
      if (e != i0 && gs[e] > gs[i1]) i1 = e;
    float s0 = gs[i0], s1 = gs[i1];
    float rn = 1.0f / (s0 + s1 + 1e-8f);
    top_idx[b * 2 + 0] = i0;
    top_idx[b * 2 + 1] = i1;
    top_w[b * 2 + 0] = s0 * rn;
    top_w[b * 2 + 1] = s1 * rn;
    atomicAdd(&counts[i0], 1);
    atomicAdd(&counts[i1], 1);
  }
}

// ---------------------------------------------------------------------------
// Kernel 2: exclusive scan of the 8 counts (single thread, trivial).
// ---------------------------------------------------------------------------
__global__ void prep_kernel(const int* __restrict__ counts,
                            int* __restrict__ offsets) {
  if (threadIdx.x == 0 && blockIdx.x == 0) {
    int s = 0;
    for (int e = 0; e < EE; ++e) { offsets[e] = s; s += counts[e]; }
    offsets[EE] = s;
  }
}

// ---------------------------------------------------------------------------
// Kernel 3: scatter tokens into per-expert lists; record inverse slot map.
// ---------------------------------------------------------------------------
__global__ __launch_bounds__(256) void scatter_kernel(
    const int* __restrict__ top_idx, const float* __restrict__ top_w,
    const int* __restrict__ offsets, int* __restrict__ cursor,
    int* __restrict__ tok_list, float* __restrict__ w_list,
    int* __restrict__ slot_of) {
  int t = blockIdx.x * blockDim.x + threadIdx.x;
  if (t >= BB) return;
#pragma unroll
  for (int k = 0; k < KKE; ++k) {
    int e = top_idx[t * 2 + k];
    float w = top_w[t * 2 + k];
    int pos = atomicAdd(&cursor[e], 1);
    int slot = offsets[e] + pos;
    tok_list[slot] = t;
    w_list[slot] = w;
    slot_of[t * 2 + k] = slot;
  }
}

// ---------------------------------------------------------------------------
// Kernel 4: fused expert FFN (relu(x@W1+b1)@W2+b2, weighted).
// Block = (16-token tile, expert). 256 threads = 8 waves.
// If stage != nullptr: coalesced stores into per-slot staging buffer.
// Else: atomicAdd fallback into out (exact: 2 commutative adds from zero).
// ---------------------------------------------------------------------------
#define DC 256     // D staging chunk
#define HCK 512    // H chunk held in LDS
#define XS 260     // padded LDS row stride for x chunk (float4-aligned)
#define HS 516     // padded LDS row stride for h chunk

__global__ __launch_bounds__(256) void ffn_kernel(
    const float* __restrict__ x, const float* __restrict__ W1,
    const float* __restrict__ b1, const float* __restrict__ W2,
    const float* __restrict__ b2, const int* __restrict__ counts,
    const int* __restrict__ offsets, const int* __restrict__ tok_list,
    const float* __restrict__ w_list, float* __restrict__ out,
    float* __restrict__ stage) {
  __shared__ float xs[16 * XS];   // 16 x 256 x-chunk (padded)
  __shared__ float hs[16 * HS];   // 16 x 512 h-chunk (padded)
  __shared__ int tok_s[16];
  __shared__ float w_s[16];

  const int e = blockIdx.y;
  const int cnt = counts[e];
  const int base = blockIdx.x * 16;
  if (base >= cnt) return;
  const int off = offsets[e];

  const int tid = threadIdx.x;
  if (tid < 16) {
    int s = base + tid;
    if (s < cnt) {
      tok_s[tid] = tok_list[off + s];
      w_s[tid] = w_list[off + s];
    } else {
      tok_s[tid] = 0;
      w_s[tid] = 0.0f;
    }
  }
  __syncthreads();

  const int wv = tid >> 5;
  const int lane = tid & 31;
  const int m = lane & 15;        // M (A row) / N (B,C col) per WMMA layout
  const int grp = lane >> 4;      // lane-group selects the K pair
  const int ksel = grp * 2;

  const float* W1e = W1 + (size_t)e * DD * HH;
  const float* W2e = W2 + (size_t)e * HH * CC;

  v8f oacc[8];
#pragma unroll
  for (int i = 0; i < 8; ++i) oacc[i] = (v8f)0.0f;

  for (int hc = 0; hc < HH / HCK; ++hc) {
    v8f hacc[4];
#pragma unroll
    for (int i = 0; i < 4; ++i) hacc[i] = (v8f)0.0f;

    for (int dc = 0; dc < DD / DC; ++dc) {
      __syncthreads();   // xs reuse / hs reads of previous chunk done
      const int dbase = dc * DC;
      // cooperative stage of x tile [16 x 256] via float4
      for (int i = tid; i < 16 * (DC / 4); i += 256) {
        int mm = i >> 6;          // DC/4 == 64 vec4 per row
        int kv = i & 63;
        float4 v = *(const float4*)(x + (size_t)tok_s[mm] * DD + dbase + kv * 4);
        *(float4*)&xs[mm * XS + kv * 4] = v;
      }
      __syncthreads();
#pragma unroll 4
      for (int kk = 0; kk < DC / 4; ++kk) {
        v2f a = *(const v2f*)&xs[m * XS + kk * 4 + ksel];
        const float* bp = W1e + (size_t)(dbase + kk * 4 + ksel) * HH +
                          hc * HCK + wv * 64 + m;
#pragma unroll
        for (int ct = 0; ct < 4; ++ct) {
          v2f bf;
          bf.x = bp[ct * 16];
          bf.y = bp[ct * 16 + HH];
          hacc[ct] = wmma_f32(a, bf, hacc[ct]);
        }
      }
    }

    // bias + relu + spill h chunk to LDS (C-layout: elem r -> row r+8*grp)
#pragma unroll
    for (int ct = 0; ct < 4; ++ct) {
      int cloc = wv * 64 + ct * 16 + m;      // column within chunk
      float bv = b1[(size_t)e * HH + hc * HCK + cloc];
#pragma unroll
      for (int r = 0; r < 8; ++r) {
        float v = hacc[ct][r] + bv;
        hs[(r + 8 * grp) * HS + cloc] = v > 0.0f ? v : 0.0f;
      }
    }
    __syncthreads();

    // second GEMM over this h chunk
#pragma unroll 2
    for (int kk = 0; kk < HCK / 4; ++kk) {
      v2f a = *(const v2f*)&hs[m * HS + kk * 4 + ksel];
      int krow = hc * HCK + kk * 4 + ksel;
#pragma unroll
      for (int oc = 0; oc < 8; ++oc) {
        int octg = wv * 8 + oc;              // output col-tile 0..63 (63 unused)
        if (octg < 63) {
          int col = octg * 16 + m;
          int colc = col < CC ? col : (CC - 1);  // clamp tail loads
          const float* bp = W2e + (size_t)krow * CC + colc;
          v2f bf;
          bf.x = bp[0];
          bf.y = bp[CC];
          oacc[oc] = wmma_f32(a, bf, oacc[oc]);
        }
      }
    }
  }

  // add b2, weight, then either stage (coalesced stores) or atomic-accumulate
#pragma unroll
  for (int oc = 0; oc < 8; ++oc) {
    int octg = wv * 8 + oc;
    if (octg < 63) {
      int col = octg * 16 + m;
      if (col < CC) {
        float b2v = b2[(size_t)e * CC + col];
        if (stage) {
#pragma unroll
          for (int r = 0; r < 8; ++r) {
            int mm = r + 8 * grp;
            if (base + mm < cnt)
              stage[(size_t)(off + base + mm) * CC + col] =
                  w_s[mm] * (oacc[oc][r] + b2v);
          }
        } else {
#pragma unroll
          for (int r = 0; r < 8; ++r) {
            int mm = r + 8 * grp;
            if (base + mm < cnt)
              atomicAdd(out + (size_t)tok_s[mm] * CC + col,
                        w_s[mm] * (oacc[oc][r] + b2v));
          }
        }
      }
    }
  }
}

// ---------------------------------------------------------------------------
// Kernel 4b: combine staged per-slot outputs: out[t] = stage[s0] + stage[s1].
// Fixed add order -> bit-deterministic, independent of slot permutation.
// ---------------------------------------------------------------------------
__global__ __launch_bounds__(256) void combine_kernel(
    const float* __restrict__ stage, const int* __restrict__ slot_of,
    float* __restrict__ out) {
  const int t = blockIdx.x;
  const int s0 = slot_of[t * 2 + 0];
  const int s1 = slot_of[t * 2 + 1];
  const float* r0 = stage + (size_t)s0 * CC;
  const float* r1 = stage + (size_t)s1 * CC;
  float* o = out + (size_t)t * CC;
  for (int c = threadIdx.x; c < CC; c += 256) o[c] = r0[c] + r1[c];
}

// ---------------------------------------------------------------------------
// Kernel 5: deterministic aux-loss reduction.
// ---------------------------------------------------------------------------
__global__ __launch_bounds__(256) void aux_kernel(
    const float* __restrict__ gs, const int* __restrict__ counts,
    float* __restrict__ aux_out) {
  __shared__ float red[EE * 256];
  const int tid = threadIdx.x;
  float s[EE];
#pragma unroll
  for (int e = 0; e < EE; ++e) s[e] = 0.0f;
  for (int b = tid; b < BB; b += 256) {
#pragma unroll
    for (int e = 0; e < EE; ++e) s[e] += gs[(size_t)b * EE + e];
  }
#pragma unroll
  for (int e = 0; e < EE; ++e) red[e * 256 + tid] = s[e];
  __syncthreads();
  for (int o = 128; o > 0; o >>= 1) {
    if (tid < o) {
#pragma unroll
      for (int e = 0; e < EE; ++e)
        red[e * 256 + tid] += red[e * 256 + tid + o];
    }
    __syncthreads();
  }
  if (tid == 0) {
    float a = 0.0f;
    for (int e = 0; e < EE; ++e) {
      float f = (float)counts[e] / (float)BB;
      float P = red[e * 256] / (float)BB;
      a += f * P;
    }
    aux_out[0] = (float)EE * a;
  }
}

// ---------------------------------------------------------------------------
extern "C" void kernel_launch(void* const* d_in, const int* in_sizes, int n_in,
                              void* d_out, int out_size, void* d_ws,
                              size_t ws_size, hipStream_t stream) {
  (void)in_sizes; (void)n_in; (void)out_size;
  const float* x  = (const float*)d_in[0];
  const float* Wg = (const float*)d_in[1];
  const float* bg = (const float*)d_in[2];
  const float* W1 = (const float*)d_in[3];
  const float* b1 = (const float*)d_in[4];
  const float* W2 = (const float*)d_in[5];
  const float* b2 = (const float*)d_in[6];
  // d_in[7] = top_k, fixed to 2

  float* out = (float*)d_out;                      // [B, C]
  float* aux = out + (size_t)BB * CC;              // scalar
  float* gs  = aux + 1;                            // [B, E]

  int*   counts   = (int*)d_ws;                    // 8
  int*   offsets  = counts + 8;                    // 9
  int*   cursor   = counts + 24;                   // 8
  int*   top_idx  = counts + 48;                   // B*2 ints
  float* top_w    = (float*)(top_idx + BB * 2);    // B*2 floats
  int*   tok_list = (int*)(top_w + BB * 2);        // B*K ints
  float* w_list   = (float*)(tok_list + BB * KKE); // B*K floats
  int*   slot_of  = (int*)(w_list + BB * KKE);     // B*K ints
  float* stage    = (float*)(slot_of + BB * KKE);  // [B*K, C] (optional)

  const size_t need_staged =
      (size_t)(48 + 5 * BB * KKE) * 4 + (size_t)BB * KKE * CC * sizeof(float);
  const bool staged = ws_size >= need_staged;

  if (!staged)
    hipMemsetAsync(out, 0, (size_t)BB * CC * sizeof(float), stream);
  hipMemsetAsync(counts, 0, 48 * sizeof(int), stream);

  gate_kernel<<<BB / 8, 256, 0, stream>>>(x, Wg, bg, gs, top_idx, top_w,
                                          counts);
  prep_kernel<<<1, 32, 0, stream>>>(counts, offsets);
  scatter_kernel<<<BB / 256, 256, 0, stream>>>(top_idx, top_w, offsets, cursor,
                                               tok_list, w_list, slot_of);
  dim3 fgrid(BB / 16, EE);   // expert-major dispatch keeps one expert's
                             // weights (48MB) hot in the 192MB L2
  ffn_kernel<<<fgrid, 256, 0, stream>>>(x, W1, b1, W2, b2, counts, offsets,
                                        tok_list, w_list, out,
                                        staged ? stage : (float*)nullptr);
  if (staged)
    combine_kernel<<<BB, 256, 0, stream>>>(stage, slot_of, out);
  aux_kernel<<<1, 256, 0, stream>>>(gs, counts, aux);
}